// N3AggregationBase_21912923144704
// MI455X (gfx1250) — compile-verified
//
#include <hip/hip_runtime.h>
#include <hip/hip_bf16.h>
#include <math.h>

// N3 aggregation pipeline for MI455X (gfx1250, wave32).
// Heavy O-contraction done as per-pixel fp32 WMMA (V_WMMA_F32_16X16X4_F32):
//   out[k,c,p] = sum_o Wsum[k,o,p] * x[c, p+off(o)]  ==  per-pixel [16xO] x [Ox64] GEMM.
// All guard conditions are removed from the hot loop by zero-padding the data
// (padded images + zero-padded Wsum), so every load is unconditional.
// Scratch requirement: ~205 MB (intermediates are L2-resident on 192MB-L2 MI455X).

#define KNB  7
#define PSZ  7
#define PR   3       // PSZ/2
#define WSW  13
#define WR   6
#define ONB  169
#define OPAD 176     // O padded to multiple of 16
#define EDIM 32
#define CDIM 64
#define HDIM 128
#define WDIM 128
#define HWPX (HDIM*WDIM)
#define WPD  (WDIM+2*WR)       // 140, padded width
#define HPD  (HDIM+2*WR)       // 140, padded height
#define PHW  (WPD*HPD)         // 19600 padded pixels
#define EPSV 1e-6f

typedef __attribute__((ext_vector_type(2))) float v2f;
typedef __attribute__((ext_vector_type(8))) float v8f;

// ---------- B-offset table: boff[o] = (di*WPD + dj)*CDIM, 0 for pad ----------
__global__ void k_boff(int* __restrict__ boff) {
    int o = threadIdx.x;
    if (o >= OPAD) return;
    int v = 0;
    if (o < ONB) v = ((o / WSW) * WPD + (o % WSW)) * CDIM;
    boff[o] = v;
}

// ---------- zero-padded xe image: xep[e][HPD][WPD] ---------------------------
__global__ __launch_bounds__(256) void k_pad_xe(const float* __restrict__ xe,
                                                float* __restrict__ xep) {
    int t = blockIdx.x * 256 + threadIdx.x;
    if (t >= EDIM * PHW) return;
    int pp = t % PHW, e = t / PHW;
    int ph = pp / WPD, pw = pp % WPD;
    int oh = ph - WR, ow = pw - WR;
    float v = 0.f;
    if (oh >= 0 && oh < HDIM && ow >= 0 && ow < WDIM)
        v = xe[e * HWPX + oh * WDIM + ow];
    xep[t] = v;
}

// ---------- zero-padded transposed x: xTp[HPD*WPD][C] ------------------------
__global__ __launch_bounds__(256) void k_pad_xT(const float* __restrict__ x,
                                                float* __restrict__ xTp) {
    int t = blockIdx.x * 256 + threadIdx.x;
    if (t >= PHW * CDIM) return;
    int c = t % CDIM, pp = t / CDIM;
    int ph = pp / WPD, pw = pp % WPD;
    int oh = ph - WR, ow = pw - WR;
    float v = 0.f;
    if (oh >= 0 && oh < HDIM && ow >= 0 && ow < WDIM)
        v = x[c * HWPX + oh * WDIM + ow];
    xTp[t] = v;
}

// ---------- temp[p] = exp(boxsum7x7(log_temp)/49) ----------------------------
__global__ __launch_bounds__(256) void k_temp(const float* __restrict__ lt,
                                              float* __restrict__ temp) {
    int p = blockIdx.x * 256 + threadIdx.x;
    if (p >= HWPX) return;
    int h = p / WDIM, w = p % WDIM;
    float s = 0.f;
    for (int dh = -PR; dh <= PR; ++dh) {
        int hh = h + dh; if (hh < 0 || hh >= HDIM) continue;
        for (int dw = -PR; dw <= PR; ++dw) {
            int ww = w + dw; if (ww < 0 || ww >= WDIM) continue;
            s += lt[hh * WDIM + ww];
        }
    }
    temp[p] = __expf(s * (1.0f / (PSZ * PSZ)));
}

// ---------- diff2 [HW][OPAD]: per-candidate embedding sq-dist ----------------
__global__ __launch_bounds__(256) void k_diff2(const float* __restrict__ xep,
                                               const float* __restrict__ ye,
                                               float* __restrict__ diff2) {
    int t = blockIdx.x * 256 + threadIdx.x;
    if (t >= HWPX * OPAD) return;
    int o = t % OPAD, p = t / OPAD;
    if (o >= ONB) { diff2[t] = 0.f; return; }   // pad lanes sum as zero in box filter
    int h = p / WDIM, w = p % WDIM;
    int q = (h + o / WSW) * WPD + (w + o % WSW);   // padded image: always in bounds
    float d = 0.f;
    #pragma unroll 8
    for (int e = 0; e < EDIM; ++e) {
        float df = ye[e * HWPX + p] - xep[e * PHW + q];
        d += df * df;
    }
    diff2[t] = d;
}

// ---------- horizontal 7-tap box sum over W, layout [planes][H][W][OPAD] -----
__global__ __launch_bounds__(256) void k_hbox(const float* __restrict__ src,
                                              float* __restrict__ dst, int nplanes) {
    long t = (long)blockIdx.x * 256 + threadIdx.x;
    long total = (long)nplanes * HWPX * OPAD;
    if (t >= total) return;
    int o = (int)(t % OPAD);
    long r = t / OPAD;              // row index over [planes*H*W]
    int w = (int)(r % WDIM);
    long base = r - w;              // start of this (plane,h) line
    float s = 0.f;
    for (int dw = -PR; dw <= PR; ++dw) {
        int ww = w + dw;
        if (ww < 0 || ww >= WDIM) continue;
        s += src[(base + ww) * OPAD + o];
    }
    dst[t] = s;
}

// ---------- vertical 7-tap + negate + /temp -> logits [HW][OPAD] -------------
__global__ __launch_bounds__(256) void k_vlogits(const float* __restrict__ hb,
                                                 const float* __restrict__ temp,
                                                 float* __restrict__ logitsT) {
    int t = blockIdx.x * 256 + threadIdx.x;
    if (t >= HWPX * OPAD) return;
    int o = t % OPAD, p = t / OPAD;
    if (o >= ONB) { logitsT[t] = -__builtin_inff(); return; }  // pad -> softmax weight 0
    int h = p / WDIM, w = p % WDIM;
    float s = 0.f;
    for (int dh = -PR; dh <= PR; ++dh) {
        int hh = h + dh; if (hh < 0 || hh >= HDIM) continue;
        s += hb[(hh * WDIM + w) * OPAD + o];
    }
    logitsT[t] = -s / temp[p];
}

// ---------- K rounds of temp-softmax-with-exclusion: one wave32 per pixel ----
__global__ __launch_bounds__(256) void k_nnn(const float* __restrict__ logitsT,
                                             float* __restrict__ Wk) {
    int wid  = (blockIdx.x * 256 + threadIdx.x) >> 5;   // pixel
    int lane = threadIdx.x & 31;
    if (wid >= HWPX) return;
    const int NS = 6;                                   // ceil(192/32) slots
    float l[NS];
    #pragma unroll
    for (int j = 0; j < NS; ++j) {
        int idx = j * 32 + lane;
        l[j] = (idx < OPAD) ? logitsT[wid * OPAD + idx] : -__builtin_inff();
    }
    for (int k = 0; k < KNB; ++k) {
        float m = -__builtin_inff();
        #pragma unroll
        for (int j = 0; j < NS; ++j) m = fmaxf(m, l[j]);
        for (int s = 16; s > 0; s >>= 1) m = fmaxf(m, __shfl_xor(m, s, 32));
        float e[NS]; float sum = 0.f;
        #pragma unroll
        for (int j = 0; j < NS; ++j) { e[j] = __expf(l[j] - m); sum += e[j]; }
        for (int s = 16; s > 0; s >>= 1) sum += __shfl_xor(sum, s, 32);
        float inv = 1.0f / sum;
        #pragma unroll
        for (int j = 0; j < NS; ++j) {
            float wv = e[j] * inv;
            int idx = j * 32 + lane;
            if (idx < OPAD) Wk[(long)k * HWPX * OPAD + (long)wid * OPAD + idx] = wv;
            l[j] += __logf(fmaxf(1.0f - wv, EPSV));     // exclusion term
        }
    }
}

// ---------- vertical box on hbox(Wk) -> WsumT [HW][OPAD][8], zero-padded -----
__global__ __launch_bounds__(256) void k_vwsum(const float* __restrict__ tmp,
                                               float* __restrict__ WsumT) {
    long t = (long)blockIdx.x * 256 + threadIdx.x;
    long total = (long)8 * HWPX * OPAD;
    if (t >= total) return;
    int o = (int)(t % OPAD);
    long r = t / OPAD;
    int p = (int)(r % HWPX);
    int k = (int)(r / HWPX);
    float s = 0.f;
    if (k < KNB) {                      // k==7 slot and o-pad slots -> 0
        int h = p / WDIM, w = p % WDIM;
        for (int dh = -PR; dh <= PR; ++dh) {
            int hh = h + dh; if (hh < 0 || hh >= HDIM) continue;
            s += tmp[((long)k * HWPX + hh * WDIM + w) * OPAD + o];
        }
    }
    WsumT[((long)p * OPAD + o) * 8 + k] = s;
}

// ---------- WMMA aggregation: per pixel D[16x64] = A[16xO] x B[Ox64] ---------
// A rows 0..6 = Wsum[k,:,p] (rows 7..15 zero via zero-padded WsumT + cndmask);
// B[o,c] = padded image fetch, always in bounds.  fp32 WMMA 16x16x4, one wave
// per pixel, 43 K-chunks x 4 N-tiles = 172 WMMAs.  Output staged through LDS
// for coalesced stores.
#define LDSROW 449                      // 448 outputs + 1 pad (bank-conflict-free)
__global__ __launch_bounds__(256) void k_agg(const float* __restrict__ WsumT,
                                             const float* __restrict__ xTp,
                                             const int*   __restrict__ boff,
                                             float* __restrict__ out) {
    __shared__ float lds[8 * LDSROW];
    int wv   = threadIdx.x >> 5;                 // wave in block: 8 pixels/block
    int wid  = blockIdx.x * 8 + wv;              // pixel
    int lane = threadIdx.x & 31;
    int p = wid;
    int h = p / WDIM, w = p % WDIM;
    int lrow = lane & 15;
    int hi   = lane >> 4;            // 0: A-K rows 0/1, D rows 0..7 | 1: K 2/3, D 8..15
    bool arow_ok = (lane & 8) == 0;  // only M rows 0..7 carry data (rows 7..15 -> 0)
    v8f acc0 = {}, acc1 = {}, acc2 = {}, acc3 = {};
    const float* wsA = WsumT + (long)p * OPAD * 8 + (lrow & 7);
    const float* xb  = xTp + (h * WPD + w) * CDIM + lrow;
    const int4*  bo4 = (const int4*)boff;

    for (int oc = 0; oc < 43; ++oc) {            // 43 chunks of 4 cover O=169 (+pad zeros)
        int4 ct  = bo4[oc];                      // offsets for o = 4oc .. 4oc+3
        int off0 = hi ? ct.z : ct.x;             // lane-half selects its K rows
        int off1 = hi ? ct.w : ct.y;
        int abase = (4 * oc + 2 * hi) * 8;
        float a0 = wsA[abase];                   // unconditional: WsumT fully padded
        float a1 = wsA[abase + 8];
        a0 = arow_ok ? a0 : 0.f;                 // v_cndmask, no branch
        a1 = arow_ok ? a1 : 0.f;
        v2f A; A.x = a0; A.y = a1;
        const float* b0p = xb + off0;
        const float* b1p = xb + off1;
        #define DO_TILE(ACC, C0)                                                    \
        {                                                                           \
            v2f B; B.x = b0p[C0]; B.y = b1p[C0];                                    \
            ACC = __builtin_amdgcn_wmma_f32_16x16x4_f32(false, A, false, B,         \
                                                        (short)0, ACC, false, false); \
        }
        DO_TILE(acc0, 0)
        DO_TILE(acc1, 16)
        DO_TILE(acc2, 32)
        DO_TILE(acc3, 48)
        #undef DO_TILE
    }

    // D layout: vgpr g holds row g (lanes 0-15) / g+8 (lanes 16-31), col = lane&15.
    // Stage [7 rows x 64 cols] per pixel into LDS, then write coalesced.
    #define STORE_TILE(ACC, C0)                                                     \
    {                                                                               \
        _Pragma("unroll")                                                           \
        for (int g = 0; g < 8; ++g) {                                               \
            int row = g + 8 * hi;                                                   \
            if (row < KNB) lds[wv * LDSROW + row * CDIM + (C0) + lrow] = ACC[g];    \
        }                                                                           \
    }
    STORE_TILE(acc0, 0)
    STORE_TILE(acc1, 16)
    STORE_TILE(acc2, 32)
    STORE_TILE(acc3, 48)
    #undef STORE_TILE
    __syncthreads();

    // Cooperative coalesced store: out[kc][p0..p0+7], 8 consecutive pixels/block.
    int p0 = blockIdx.x * 8;
    for (int idx = threadIdx.x; idx < KNB * CDIM * 8; idx += 256) {
        int j  = idx & 7;                        // pixel within block
        int kc = idx >> 3;                       // k*C + c
        out[(long)kc * HWPX + p0 + j] = lds[j * LDSROW + kc];
    }
}

extern "C" void kernel_launch(void* const* d_in, const int* in_sizes, int n_in,
                              void* d_out, int out_size, void* d_ws, size_t ws_size,
                              hipStream_t stream) {
    (void)in_sizes; (void)n_in; (void)out_size; (void)ws_size;
    const float* x  = (const float*)d_in[0];   // [64,128,128]
    const float* xe = (const float*)d_in[1];   // [32,128,128]
    const float* ye = (const float*)d_in[2];   // [32,128,128]
    const float* lt = (const float*)d_in[3];   // [1,128,128]
    float* out = (float*)d_out;                // [7*64,128,128]

    // Workspace carve-up (floats). Total ~205 MB; mostly L2-resident on MI455X.
    float* ws = (float*)d_ws;
    const long N_PO  = (long)HWPX * OPAD;             //  2,883,584
    const long N_KPO = (long)KNB * HWPX * OPAD;       // 20,185,088
    const long N_WST = (long)HWPX * OPAD * 8;         // 23,068,672
    float* bufA   = ws;                               // diff2, then logitsT  (11.5 MB)
    float* bufB   = bufA + N_PO;                      // hbox scratch         (11.5 MB)
    float* temp   = bufB + N_PO;                      // [HW]                 (64 KB)
    float* xep    = temp + HWPX;                      // padded xe            (2.5 MB)
    float* xTp    = xep + (long)EDIM * PHW;           // padded x^T           (5 MB)
    int*   boff   = (int*)(xTp + (long)PHW * CDIM);   // 176 ints
    float* region = (float*)(boff + OPAD);            // Wk then WsumT        (92.3 MB)
    float* Wk     = region;
    float* WsumT  = region;                           // aliased: Wk dead before write
    float* tmp    = region + N_WST;                   // hbox(Wk)             (80.7 MB)

    const int TPB = 256;
    auto blk = [](long n) { return (unsigned)((n + 255) / 256); };

    k_boff<<<1, TPB, 0, stream>>>(boff);
    k_pad_xe<<<blk((long)EDIM * PHW), TPB, 0, stream>>>(xe, xep);
    k_pad_xT<<<blk((long)PHW * CDIM), TPB, 0, stream>>>(x, xTp);
    k_temp<<<blk(HWPX), TPB, 0, stream>>>(lt, temp);
    k_diff2<<<blk(N_PO), TPB, 0, stream>>>(xep, ye, bufA);
    k_hbox<<<blk(N_PO), TPB, 0, stream>>>(bufA, bufB, 1);
    k_vlogits<<<blk(N_PO), TPB, 0, stream>>>(bufB, temp, bufA);     // bufA := logits
    k_nnn<<<blk((long)HWPX * 32), TPB, 0, stream>>>(bufA, Wk);
    k_hbox<<<blk(N_KPO), TPB, 0, stream>>>(Wk, tmp, KNB);
    k_vwsum<<<blk((long)8 * HWPX * OPAD), TPB, 0, stream>>>(tmp, WsumT);
    k_agg<<<blk((long)HWPX * 32), TPB, 0, stream>>>(WsumT, xTp, boff, out);
}